// RVQBottleneck_34600256537524
// MI455X (gfx1250) — compile-verified
//
#include <hip/hip_runtime.h>

typedef __attribute__((ext_vector_type(2))) float v2f;
typedef __attribute__((ext_vector_type(4))) float v4f;
typedef __attribute__((ext_vector_type(8))) float v8f;

#define RVQ_B 8
#define RVQ_C 128
#define RVQ_N 4096
#define RVQ_Q 8
#define RVQ_K 2048
#define RVQ_BN (RVQ_B * RVQ_N)               // 32768 rows
#define ROWS_PER_BLK 16
#define STEP_BLOCKS (RVQ_BN / ROWS_PER_BLK)  // 2048
#define GROUPS (RVQ_K / 16)                  // 128 column groups per stage

// ---------------------------------------------------------------------------
// Transpose (B,C,N) -> (B,N,C)   (x -> residual)
// ---------------------------------------------------------------------------
__global__ __launch_bounds__(256) void rvq_transpose_in(
    const float* __restrict__ x, float* __restrict__ r) {
  __shared__ float tile[32][33];
  const int b = blockIdx.z;
  const int n0 = blockIdx.x * 32;
  const int c0 = blockIdx.y * 32;
  const int tx = threadIdx.x, ty = threadIdx.y;
  const float* xb = x + (size_t)b * RVQ_C * RVQ_N;
  float* rb = r + (size_t)b * RVQ_N * RVQ_C;
#pragma unroll
  for (int k = 0; k < 32; k += 8)
    tile[ty + k][tx] = xb[(size_t)(c0 + ty + k) * RVQ_N + n0 + tx];
  __syncthreads();
#pragma unroll
  for (int k = 0; k < 32; k += 8)
    rb[(size_t)(n0 + ty + k) * RVQ_C + c0 + tx] = tile[tx][ty + k];
}

// ---------------------------------------------------------------------------
// Transpose (B,N,C) -> (B,C,N)   (quant -> out)
// ---------------------------------------------------------------------------
__global__ __launch_bounds__(256) void rvq_transpose_out(
    const float* __restrict__ qn, float* __restrict__ out) {
  __shared__ float tile[32][33];
  const int b = blockIdx.z;
  const int n0 = blockIdx.x * 32;
  const int c0 = blockIdx.y * 32;
  const int tx = threadIdx.x, ty = threadIdx.y;
  const float* qb = qn + (size_t)b * RVQ_N * RVQ_C;
  float* ob = out + (size_t)b * RVQ_C * RVQ_N;
#pragma unroll
  for (int k = 0; k < 32; k += 8)
    tile[ty + k][tx] = qb[(size_t)(n0 + ty + k) * RVQ_C + c0 + tx];
  __syncthreads();
#pragma unroll
  for (int k = 0; k < 32; k += 8)
    ob[(size_t)(c0 + ty + k) * RVQ_N + n0 + tx] = tile[tx][ty + k];
}

// ---------------------------------------------------------------------------
// ||c||^2 for every codebook row (Q*K rows of length C)
// ---------------------------------------------------------------------------
__global__ __launch_bounds__(256) void rvq_csq(
    const float* __restrict__ cb, float* __restrict__ csq) {
  const int i = blockIdx.x * blockDim.x + threadIdx.x;  // 0 .. Q*K-1
  const v4f* row = (const v4f*)(cb + (size_t)i * RVQ_C);
  float s = 0.f;
#pragma unroll
  for (int j = 0; j < RVQ_C / 4; ++j) {
    v4f v = row[j];
    s += v.x * v.x + v.y * v.y + v.z * v.z + v.w * v.w;
  }
  csq[i] = s;
}

// ---------------------------------------------------------------------------
// Pack codebooks into WMMA-B operand order so step-kernel B fetches are
// fully coalesced b128 loads.
// pack[(((q*GROUPS + g)*16 + p)*32 + lane)] (float4) =
//   col n = g*16 + (lane&15), hi = lane>>4:
//   { cb[n][8p+2hi], cb[n][8p+2hi+1], cb[n][8p+4+2hi], cb[n][8p+4+2hi+1] }
// ---------------------------------------------------------------------------
__global__ __launch_bounds__(256) void rvq_pack_cb(
    const float* __restrict__ cb, float* __restrict__ pack) {
  const int t = blockIdx.x * 256 + threadIdx.x;  // 0 .. Q*GROUPS*16*32-1
  const int lane = t & 31;
  const int p = (t >> 5) & 15;
  const int g = (t >> 9) & (GROUPS - 1);
  const int q = t >> 16;
  const int hi = lane >> 4, m = lane & 15;
  const float* row = cb + ((size_t)q * RVQ_K + g * 16 + m) * RVQ_C;
  v2f lo = *(const v2f*)(row + 8 * p + 2 * hi);
  v2f h2 = *(const v2f*)(row + 8 * p + 4 + 2 * hi);
  v4f o;
  o.xy = lo;
  o.zw = h2;
  *(v4f*)(pack + (size_t)t * 4) = o;
}

// ---------------------------------------------------------------------------
// One RVQ stage: fused (residual x codebook^T) GEMM via WMMA f32 16x16x4,
// per-row argmin over K, codebook gather, residual/quant update, commit partial.
// Block: 256 threads (8 waves), 16 rows per block. Scheduler-pipelined B stream.
// ---------------------------------------------------------------------------
__global__ __launch_bounds__(256) void rvq_step(
    const float* __restrict__ cb_q,     // (K, C) codebook for this stage
    const float* __restrict__ pack_q,   // packed B operands for this stage
    const float* __restrict__ csq_q,    // (K,)
    float* __restrict__ residual,       // (BN, C)
    float* __restrict__ quant,          // (BN, C)
    float* __restrict__ idx_out,        // (BN, Q) as float
    float* __restrict__ commit_parts,   // (Q, STEP_BLOCKS)
    int q) {
  const int tid = threadIdx.x;
  const int wave = tid >> 5;  // 0..7
  const int lane = tid & 31;
  const int m = lane & 15;    // row (A) / col (B) within 16-group
  const int hi = lane >> 4;   // selects K sub-pair of the 4-deep chunk
  const int row_base = blockIdx.x * ROWS_PER_BLK;

  // ---- Load A tile: 16 rows x 128 (f32) into registers, WMMA 16x16x4 layout.
  const float* Arow = residual + (size_t)(row_base + m) * RVQ_C;
  v2f a[32];
#pragma unroll
  for (int ck = 0; ck < 32; ++ck)
    a[ck] = *(const v2f*)(Arow + 4 * ck + 2 * hi);

  float best[8];
  int bestIdx[8];
#pragma unroll
  for (int j = 0; j < 8; ++j) { best[j] = 3.402823466e+38f; bestIdx[j] = 0; }

  // ---- Sweep K: wave w owns groups {w, w+8, ..., w+120}; fully unrolled so
  // the scheduler software-pipelines the coalesced b128 clauses across groups.
  const float* gbase = pack_q + (size_t)wave * (16 * 32 * 4) + lane * 4;
#pragma unroll
  for (int it = 0; it < GROUPS / 8; ++it) {
    const float* base = gbase + (size_t)it * (8 * 16 * 32 * 4);
    v4f bq[16];
#pragma unroll
    for (int p = 0; p < 16; ++p) bq[p] = *(const v4f*)(base + p * 128);

    v8f acc0 = {0.f, 0.f, 0.f, 0.f, 0.f, 0.f, 0.f, 0.f};
    v8f acc1 = {0.f, 0.f, 0.f, 0.f, 0.f, 0.f, 0.f, 0.f};
#pragma unroll
    for (int p = 0; p < 16; ++p) {
      acc0 = __builtin_amdgcn_wmma_f32_16x16x4_f32(
          false, a[2 * p], false, bq[p].xy, (short)0, acc0, false, false);
      acc1 = __builtin_amdgcn_wmma_f32_16x16x4_f32(
          false, a[2 * p + 1], false, bq[p].zw, (short)0, acc1, false, false);
    }
    // D layout: lanes 0-15 -> N=lane, M=j ; lanes 16-31 -> N=lane-16, M=j+8.
    const int n = (it * 8 + wave) * 16 + m;
    const float cs = csq_q[n];
#pragma unroll
    for (int j = 0; j < 8; ++j) {
      float s = cs - 2.0f * (acc0[j] + acc1[j]);
      if (s < best[j] || (s == best[j] && n < bestIdx[j])) {
        best[j] = s;
        bestIdx[j] = n;
      }
    }
  }

  // ---- Reduce argmin across the 16 lanes of each half-wave.
#pragma unroll
  for (int off = 1; off < 16; off <<= 1) {
#pragma unroll
    for (int j = 0; j < 8; ++j) {
      float ov = __shfl_xor(best[j], off, 32);
      int oi = __shfl_xor(bestIdx[j], off, 32);
      if (ov < best[j] || (ov == best[j] && oi < bestIdx[j])) {
        best[j] = ov;
        bestIdx[j] = oi;
      }
    }
  }

  __shared__ float s_val[8][16];
  __shared__ int s_idx[8][16];
  __shared__ int s_final[16];
  __shared__ float s_part[8];
  if (m == 0) {
#pragma unroll
    for (int j = 0; j < 8; ++j) {
      s_val[wave][j + 8 * hi] = best[j];
      s_idx[wave][j + 8 * hi] = bestIdx[j];
    }
  }
  __syncthreads();

  // ---- Reduce across the 8 waves (each covered disjoint columns).
  if (tid < 16) {
    float bv = s_val[0][tid];
    int bi = s_idx[0][tid];
#pragma unroll
    for (int w = 1; w < 8; ++w) {
      float v = s_val[w][tid];
      int i = s_idx[w][tid];
      if (v < bv || (v == bv && i < bi)) { bv = v; bi = i; }
    }
    s_final[tid] = bi;
    idx_out[(size_t)(row_base + tid) * RVQ_Q + q] = (float)bi;
  }
  __syncthreads();

  // ---- Update phase: gather codebook rows, update residual & quant,
  //      accumulate commit = sum (cb - r)^2 over the tile.
  const int c_ = tid & 127;  // coalesced along C for residual/quant
  const int r_ = tid >> 7;   // 0..1
  float lsum = 0.f;
#pragma unroll
  for (int rr = 0; rr < ROWS_PER_BLK; rr += 2) {
    const int row = rr + r_;
    const int bi = s_final[row];
    const float cv = cb_q[(size_t)bi * RVQ_C + c_];
    const size_t off = (size_t)(row_base + row) * RVQ_C + c_;
    const float rold = residual[off];
    const float diff = cv - rold;
    lsum += diff * diff;
    residual[off] = -diff;  // r - cb
    quant[off] += cv;
  }
  // deterministic block reduction of the commit partial
#pragma unroll
  for (int off = 16; off >= 1; off >>= 1) lsum += __shfl_xor(lsum, off, 32);
  if (lane == 0) s_part[wave] = lsum;
  __syncthreads();
  if (tid == 0) {
    float t = 0.f;
#pragma unroll
    for (int w = 0; w < 8; ++w) t += s_part[w];
    commit_parts[(size_t)q * STEP_BLOCKS + blockIdx.x] = t;
  }
}

// ---------------------------------------------------------------------------
// Final loss: mean over Q of (sum of squared diffs / (B*N*C))
// ---------------------------------------------------------------------------
__global__ __launch_bounds__(256) void rvq_loss(
    const float* __restrict__ parts, float* __restrict__ loss) {
  __shared__ float sm[256];
  float s = 0.f;
  for (int i = threadIdx.x; i < RVQ_Q * STEP_BLOCKS; i += 256) s += parts[i];
  sm[threadIdx.x] = s;
  __syncthreads();
  for (int o = 128; o > 0; o >>= 1) {
    if (threadIdx.x < o) sm[threadIdx.x] += sm[threadIdx.x + o];
    __syncthreads();
  }
  if (threadIdx.x == 0)
    *loss = sm[0] * (1.0f / ((float)RVQ_Q * (float)RVQ_B * (float)RVQ_N * (float)RVQ_C));
}

// ---------------------------------------------------------------------------
extern "C" void kernel_launch(void* const* d_in, const int* in_sizes, int n_in,
                              void* d_out, int out_size, void* d_ws, size_t ws_size,
                              hipStream_t stream) {
  const float* x = (const float*)d_in[0];    // (B, C, N)
  const float* cbs = (const float*)d_in[1];  // (Q, K, C)

  float* out = (float*)d_out;                            // (B, C, N)
  float* idx_out = out + (size_t)RVQ_B * RVQ_C * RVQ_N;  // (B, N, Q) as float
  float* loss_out = idx_out + (size_t)RVQ_B * RVQ_N * RVQ_Q;

  float* residual = (float*)d_ws;                    // (BN, C)
  float* quant = residual + (size_t)RVQ_BN * RVQ_C;  // (BN, C)
  float* csq = quant + (size_t)RVQ_BN * RVQ_C;       // (Q*K,)
  float* parts = csq + (size_t)RVQ_Q * RVQ_K;        // (Q, STEP_BLOCKS)
  float* pack = parts + (size_t)RVQ_Q * STEP_BLOCKS; // (Q*K*C,) packed B

  hipMemsetAsync(quant, 0, (size_t)RVQ_BN * RVQ_C * sizeof(float), stream);

  dim3 tgrid(RVQ_N / 32, RVQ_C / 32, RVQ_B);
  dim3 tblk(32, 8);
  rvq_transpose_in<<<tgrid, tblk, 0, stream>>>(x, residual);
  rvq_csq<<<(RVQ_Q * RVQ_K) / 256, 256, 0, stream>>>(cbs, csq);
  rvq_pack_cb<<<(RVQ_Q * GROUPS * 16 * 32) / 256, 256, 0, stream>>>(cbs, pack);

  for (int q = 0; q < RVQ_Q; ++q) {
    rvq_step<<<STEP_BLOCKS, 256, 0, stream>>>(
        cbs + (size_t)q * RVQ_K * RVQ_C, pack + (size_t)q * RVQ_K * RVQ_C,
        csq + (size_t)q * RVQ_K, residual, quant, idx_out, parts, q);
  }

  rvq_transpose_out<<<tgrid, tblk, 0, stream>>>(quant, out);
  rvq_loss<<<1, 256, 0, stream>>>(parts, loss_out);
}